// SRGCN_37005438222454
// MI455X (gfx1250) — compile-verified
//
#include <hip/hip_runtime.h>
#include <hip/hip_bf16.h>
#include <math.h>

// Problem constants (from reference)
#define Bb 16
#define Tt 36
#define Nn 1024
#define CIN 64
#define COUT 64
#define BT (Bb*Tt)

typedef float v2f __attribute__((ext_vector_type(2)));
typedef float v8f __attribute__((ext_vector_type(8)));

__device__ __forceinline__ v8f wmma_f32(v2f a, v2f b, v8f c) {
  // 8 args: (neg_a, A, neg_b, B, c_mod, C, reuse_a, reuse_b)
  return __builtin_amdgcn_wmma_f32_16x16x4_f32(false, a, false, b, (short)0, c, false, false);
}

// ---------------------------------------------------------------------------
// K1: Z = s_w @ lin_w^T + lin_b  -> Wb.  Wave computes a 16x64 strip:
// 1 A-fragment feeds 4 B-fragments (5 loads / 4 WMMA).
// ---------------------------------------------------------------------------
__global__ __launch_bounds__(256) void srgcn_k1_lin_gemm(
    const float* __restrict__ sw, const float* __restrict__ lw,
    const float* __restrict__ lb, float* __restrict__ Wb) {
  int wave = threadIdx.x >> 5;
  int lane = threadIdx.x & 31;
  int hl = lane >> 4;          // lane half (K split)
  int r  = lane & 15;          // A row / B-D col within tile
  int strip = blockIdx.x * 8 + wave;      // 64 row-tiles x 16 col-strips
  int i0 = (strip >> 4) * 16;
  int j0 = (strip & 15) * 64;
  const float* arow = sw + (size_t)(i0 + r) * Nn;
  const float* brow0 = lw + (size_t)(j0 +  0 + r) * Nn;  // B[k,j]=lin_w[j,k]
  const float* brow1 = lw + (size_t)(j0 + 16 + r) * Nn;
  const float* brow2 = lw + (size_t)(j0 + 32 + r) * Nn;
  const float* brow3 = lw + (size_t)(j0 + 48 + r) * Nn;
  v8f acc0 = {}, acc1 = {}, acc2 = {}, acc3 = {};
#pragma unroll 2
  for (int k = 0; k < Nn; k += 4) {
    v2f a, b;
    a.x = arow[k + 2*hl]; a.y = arow[k + 2*hl + 1];
    b.x = brow0[k + 2*hl]; b.y = brow0[k + 2*hl + 1];
    acc0 = wmma_f32(a, b, acc0);
    b.x = brow1[k + 2*hl]; b.y = brow1[k + 2*hl + 1];
    acc1 = wmma_f32(a, b, acc1);
    b.x = brow2[k + 2*hl]; b.y = brow2[k + 2*hl + 1];
    acc2 = wmma_f32(a, b, acc2);
    b.x = brow3[k + 2*hl]; b.y = brow3[k + 2*hl + 1];
    acc3 = wmma_f32(a, b, acc3);
  }
  v8f* accs[4] = {&acc0, &acc1, &acc2, &acc3};
#pragma unroll
  for (int ct = 0; ct < 4; ++ct) {
    float bias = lb[j0 + ct*16 + r];
#pragma unroll
    for (int v = 0; v < 8; ++v)
      Wb[(size_t)(i0 + v + 8*hl) * Nn + j0 + ct*16 + r] = (*accs[ct])[v] + bias;
  }
}

// ---------------------------------------------------------------------------
// K2: in-place row softmax of Wb (one 256-thread block per row)
// ---------------------------------------------------------------------------
__global__ __launch_bounds__(256) void srgcn_k2_softmax(float* __restrict__ Wb) {
  __shared__ float red[256];
  float* p = Wb + (size_t)blockIdx.x * Nn;
  int t = threadIdx.x;
  float m = -1e30f;
  for (int j = t; j < Nn; j += 256) m = fmaxf(m, p[j]);
  red[t] = m; __syncthreads();
  for (int s = 128; s > 0; s >>= 1) { if (t < s) red[t] = fmaxf(red[t], red[t+s]); __syncthreads(); }
  m = red[0]; __syncthreads();
  float sum = 0.f;
  for (int j = t; j < Nn; j += 256) { float e = __expf(p[j] - m); p[j] = e; sum += e; }
  red[t] = sum; __syncthreads();
  for (int s = 128; s > 0; s >>= 1) { if (t < s) red[t] += red[t+s]; __syncthreads(); }
  float inv = 1.0f / red[0];
  for (int j = t; j < Nn; j += 256) p[j] *= inv;
}

// ---------------------------------------------------------------------------
// K3: lambda_max of L = I - W via power iteration + Rayleigh quotient.
// ---------------------------------------------------------------------------
__global__ __launch_bounds__(1024) void srgcn_k3_lambda(
    const float* __restrict__ Wb, float* __restrict__ lam_out) {
  __shared__ float v[Nn];
  __shared__ float red[Nn];
  int n = threadIdx.x;
  unsigned h = n * 2654435761u + 12345u; h ^= h >> 13; h *= 2246822519u; h ^= h >> 16;
  v[n] = (float)(h & 0xFFFF) / 65536.0f - 0.5f + 1e-3f;
  __syncthreads();
  const float* wr = Wb + (size_t)n * Nn;
  for (int it = 0; it < 24; ++it) {
    float s = 0.f;
    for (int m = 0; m < Nn; ++m) s += wr[m] * v[m];
    float y = v[n] - s;                 // (L v)[n]
    red[n] = y * y; __syncthreads();
    for (int st = 512; st > 0; st >>= 1) { if (n < st) red[n] += red[n+st]; __syncthreads(); }
    float inv = rsqrtf(red[0] + 1e-30f);
    __syncthreads();
    v[n] = y * inv;
    __syncthreads();
  }
  float s = 0.f;
  for (int m = 0; m < Nn; ++m) s += wr[m] * v[m];
  float y = v[n] - s;
  red[n] = v[n] * y; __syncthreads();   // ||v||=1 -> Rayleigh = v.Lv
  for (int st = 512; st > 0; st >>= 1) { if (n < st) red[n] += red[n+st]; __syncthreads(); }
  if (n == 0) lam_out[0] = fabsf(red[0]) + 1e-12f;
}

// ---------------------------------------------------------------------------
// K4: elementwise Chebyshev, stored TRANSPOSED for row-major WMMA A loads.
// ---------------------------------------------------------------------------
__global__ __launch_bounds__(256) void srgcn_k4_cheb(
    const float* __restrict__ Wb, const float* __restrict__ lam_p,
    float* __restrict__ WT, float* __restrict__ LtT, float* __restrict__ T2T) {
  int idx = blockIdx.x * 256 + threadIdx.x;   // idx = n*N + m (write-coalesced)
  int n = idx >> 10, m = idx & (Nn - 1);
  float scale = 2.0f / lam_p[0];
  float w = Wb[(size_t)m * Nn + n];
  float eye = (m == n) ? 1.f : 0.f;
  float lt = scale * (eye - w) - eye;
  WT[idx]  = w;
  LtT[idx] = lt;
  T2T[idx] = 2.f * lt * lt - eye;
}

// ---------------------------------------------------------------------------
// K5: c[n] = a0 + a1*colsum(Lt)[n] + a2*colsum(T2)[n]
// ---------------------------------------------------------------------------
__global__ __launch_bounds__(256) void srgcn_k5_csum(
    const float* __restrict__ LtT, const float* __restrict__ T2T,
    const float* __restrict__ alpha, float* __restrict__ cbuf) {
  int n = blockIdx.x * 256 + threadIdx.x;
  const float* r1 = LtT + (size_t)n * Nn;
  const float* r2 = T2T + (size_t)n * Nn;
  float s1 = 0.f, s2 = 0.f;
  for (int m = 0; m < Nn; ++m) { s1 += r1[m]; s2 += r2[m]; }
  cbuf[n] = alpha[0] + alpha[1] * s1 + alpha[2] * s2;
}

// ---------------------------------------------------------------------------
// K6: per (bt, 256-row block): each wave owns 32 rows.
// Stage 1: wx = W^T x (8 accs, 6 loads / 8 WMMA) -> per-wave LDS slice.
// Stage 2: s_r_k = x@beta_k + wx@theta_k; wx A-frags come from LDS (ds_load).
// k=0 -> d_out (pre-relu), k=1,2 -> workspace.
// ---------------------------------------------------------------------------
__global__ __launch_bounds__(256) void srgcn_k6_sr(
    const float* __restrict__ WT, const float* __restrict__ x,
    const float* __restrict__ beta, const float* __restrict__ theta,
    float* __restrict__ sr1, float* __restrict__ sr2, float* __restrict__ out) {
  __shared__ float wx_s[256 * 64];            // 64 KB, per-wave 32x64 slices
  int wave = threadIdx.x >> 5;
  int lane = threadIdx.x & 31;
  int hl = lane >> 4, r = lane & 15;
  int bt = blockIdx.y;
  int nbase = blockIdx.x * 256 + wave * 32;
  const float* xbt = x + (size_t)bt * Nn * CIN;

  // ---- Stage 1: wx rows [nbase, nbase+32), all 64 cols ----
  {
    const float* a0 = WT + (size_t)(nbase + r) * Nn;
    const float* a1 = WT + (size_t)(nbase + 16 + r) * Nn;
    v8f acc[2][4] = {};
#pragma unroll 2
    for (int k = 0; k < Nn; k += 4) {
      v2f af0, af1;
      af0.x = a0[k + 2*hl]; af0.y = a0[k + 2*hl + 1];
      af1.x = a1[k + 2*hl]; af1.y = a1[k + 2*hl + 1];
      const float* br0 = xbt + (size_t)(k + 2*hl) * CIN;
      const float* br1 = xbt + (size_t)(k + 2*hl + 1) * CIN;
#pragma unroll
      for (int ct = 0; ct < 4; ++ct) {
        v2f bf; bf.x = br0[ct*16 + r]; bf.y = br1[ct*16 + r];
        acc[0][ct] = wmma_f32(af0, bf, acc[0][ct]);
        acc[1][ct] = wmma_f32(af1, bf, acc[1][ct]);
      }
    }
    float* wxw = wx_s + wave * 32 * 64;
#pragma unroll
    for (int rt = 0; rt < 2; ++rt)
#pragma unroll
      for (int ct = 0; ct < 4; ++ct)
#pragma unroll
        for (int v = 0; v < 8; ++v)
          wxw[(rt*16 + v + 8*hl) * 64 + ct*16 + r] = acc[rt][ct][v];
  }
  // No barrier: each wave reads only its own LDS slice; LDS ops are in-order.

  // ---- Stage 2: s_r_k rows ----
  int b_ = bt / Tt, t_ = bt % Tt;
  const float* xr0 = xbt + (size_t)(nbase + r) * CIN;
  const float* xr1 = xbt + (size_t)(nbase + 16 + r) * CIN;
  const float* wr0 = wx_s + wave * 32 * 64 + r * 64;
  const float* wr1 = wx_s + wave * 32 * 64 + (16 + r) * 64;
  for (int ko = 0; ko < 3; ++ko) {
    const float* Bk = beta  + ko * (CIN * COUT);
    const float* Hk = theta + ko * (CIN * COUT);
    v8f d[2][4] = {};
#pragma unroll 2
    for (int k = 0; k < CIN; k += 4) {
      v2f ax0, ax1, aw0, aw1;
      ax0.x = xr0[k + 2*hl]; ax0.y = xr0[k + 2*hl + 1];
      ax1.x = xr1[k + 2*hl]; ax1.y = xr1[k + 2*hl + 1];
      aw0.x = wr0[k + 2*hl]; aw0.y = wr0[k + 2*hl + 1];  // ds_load
      aw1.x = wr1[k + 2*hl]; aw1.y = wr1[k + 2*hl + 1];
      const float* Br0 = Bk + (k + 2*hl) * COUT;
      const float* Br1 = Bk + (k + 2*hl + 1) * COUT;
      const float* Hr0 = Hk + (k + 2*hl) * COUT;
      const float* Hr1 = Hk + (k + 2*hl + 1) * COUT;
#pragma unroll
      for (int ct = 0; ct < 4; ++ct) {
        v2f bB; bB.x = Br0[ct*16 + r]; bB.y = Br1[ct*16 + r];
        d[0][ct] = wmma_f32(ax0, bB, d[0][ct]);
        d[1][ct] = wmma_f32(ax1, bB, d[1][ct]);
        v2f bH; bH.x = Hr0[ct*16 + r]; bH.y = Hr1[ct*16 + r];
        d[0][ct] = wmma_f32(aw0, bH, d[0][ct]);
        d[1][ct] = wmma_f32(aw1, bH, d[1][ct]);
      }
    }
    if (ko == 0) {
#pragma unroll
      for (int rt = 0; rt < 2; ++rt)
#pragma unroll
        for (int ct = 0; ct < 4; ++ct)
#pragma unroll
          for (int v = 0; v < 8; ++v) {
            int n = nbase + rt*16 + v + 8*hl, o = ct*16 + r;
            out[(((size_t)b_ * Nn + n) * COUT + o) * Tt + t_] = d[rt][ct][v];
          }
    } else {
      float* srp = (ko == 1 ? sr1 : sr2) + (size_t)bt * Nn * COUT;
#pragma unroll
      for (int rt = 0; rt < 2; ++rt)
#pragma unroll
        for (int ct = 0; ct < 4; ++ct)
#pragma unroll
          for (int v = 0; v < 8; ++v)
            srp[(size_t)(nbase + rt*16 + v + 8*hl) * COUT + ct*16 + r] = d[rt][ct][v];
    }
  }
}

// ---------------------------------------------------------------------------
// K7: out[n,o] = relu( s_r0 (in d_out) + c[n]
//                    + sum_m LtT[n,m]*sr1[m,o] + sum_m T2T[n,m]*sr2[m,o] )
// Wave computes 32x64: 8 shared accumulators for BOTH matrices
// (12 loads / 16 WMMA per k-step).
// ---------------------------------------------------------------------------
__global__ __launch_bounds__(256) void srgcn_k7_out(
    const float* __restrict__ LtT, const float* __restrict__ T2T,
    const float* __restrict__ sr1, const float* __restrict__ sr2,
    const float* __restrict__ cbuf, float* __restrict__ out) {
  int wave = threadIdx.x >> 5;
  int lane = threadIdx.x & 31;
  int hl = lane >> 4, r = lane & 15;
  int bt = blockIdx.y;
  int nbase = blockIdx.x * 256 + wave * 32;
  const float* l0 = LtT + (size_t)(nbase + r) * Nn;
  const float* l1 = LtT + (size_t)(nbase + 16 + r) * Nn;
  const float* q0 = T2T + (size_t)(nbase + r) * Nn;
  const float* q1 = T2T + (size_t)(nbase + 16 + r) * Nn;
  const float* s1p = sr1 + (size_t)bt * Nn * COUT;
  const float* s2p = sr2 + (size_t)bt * Nn * COUT;
  v8f acc[2][4] = {};
#pragma unroll 2
  for (int m = 0; m < Nn; m += 4) {
    v2f al0, al1, aq0, aq1;
    al0.x = l0[m + 2*hl]; al0.y = l0[m + 2*hl + 1];
    al1.x = l1[m + 2*hl]; al1.y = l1[m + 2*hl + 1];
    aq0.x = q0[m + 2*hl]; aq0.y = q0[m + 2*hl + 1];
    aq1.x = q1[m + 2*hl]; aq1.y = q1[m + 2*hl + 1];
    const float* b1r0 = s1p + (size_t)(m + 2*hl) * COUT;
    const float* b1r1 = s1p + (size_t)(m + 2*hl + 1) * COUT;
    const float* b2r0 = s2p + (size_t)(m + 2*hl) * COUT;
    const float* b2r1 = s2p + (size_t)(m + 2*hl + 1) * COUT;
#pragma unroll
    for (int ct = 0; ct < 4; ++ct) {
      v2f b1; b1.x = b1r0[ct*16 + r]; b1.y = b1r1[ct*16 + r];
      acc[0][ct] = wmma_f32(al0, b1, acc[0][ct]);
      acc[1][ct] = wmma_f32(al1, b1, acc[1][ct]);
      v2f b2; b2.x = b2r0[ct*16 + r]; b2.y = b2r1[ct*16 + r];
      acc[0][ct] = wmma_f32(aq0, b2, acc[0][ct]);
      acc[1][ct] = wmma_f32(aq1, b2, acc[1][ct]);
    }
  }
  int b_ = bt / Tt, t_ = bt % Tt;
#pragma unroll
  for (int rt = 0; rt < 2; ++rt)
#pragma unroll
    for (int ct = 0; ct < 4; ++ct)
#pragma unroll
      for (int v = 0; v < 8; ++v) {
        int n = nbase + rt*16 + v + 8*hl, o = ct*16 + r;
        size_t oi = (((size_t)b_ * Nn + n) * COUT + o) * Tt + t_;
        float val = out[oi] + cbuf[n] + acc[rt][ct][v];
        out[oi] = fmaxf(val, 0.f);
      }
}

// ---------------------------------------------------------------------------
extern "C" void kernel_launch(void* const* d_in, const int* in_sizes, int n_in,
                              void* d_out, int out_size, void* d_ws, size_t ws_size,
                              hipStream_t stream) {
  (void)in_sizes; (void)n_in; (void)out_size; (void)ws_size;
  const float* s_w   = (const float*)d_in[0];
  const float* x     = (const float*)d_in[1];
  const float* lin_w = (const float*)d_in[2];
  const float* lin_b = (const float*)d_in[3];
  const float* alpha = (const float*)d_in[4];
  const float* beta  = (const float*)d_in[5];
  const float* theta = (const float*)d_in[6];
  float* out = (float*)d_out;

  // Workspace (floats): W, WT, LtT, T2T (N*N each), c (N), lam(pad 64),
  // sr1, sr2 (BT*N*COUT each). Total ~319 MB.
  float* ws  = (float*)d_ws;
  size_t NN = (size_t)Nn * Nn;
  float* Wb   = ws;
  float* WT   = Wb  + NN;
  float* LtT  = WT  + NN;
  float* T2T  = LtT + NN;
  float* cbuf = T2T + NN;
  float* lam  = cbuf + Nn;
  float* sr1  = lam + 64;
  float* sr2  = sr1 + (size_t)BT * Nn * COUT;

  srgcn_k1_lin_gemm<<<128, 256, 0, stream>>>(s_w, lin_w, lin_b, Wb);
  srgcn_k2_softmax<<<Nn, 256, 0, stream>>>(Wb);
  srgcn_k3_lambda<<<1, 1024, 0, stream>>>(Wb, lam);
  srgcn_k4_cheb<<<(int)(NN / 256), 256, 0, stream>>>(Wb, lam, WT, LtT, T2T);
  srgcn_k5_csum<<<Nn / 256, 256, 0, stream>>>(LtT, T2T, alpha, cbuf);
  srgcn_k6_sr<<<dim3(Nn / 256, BT), 256, 0, stream>>>(WT, x, beta, theta, sr1, sr2, out);
  srgcn_k7_out<<<dim3(Nn / 256, BT), 256, 0, stream>>>(LtT, T2T, sr1, sr2, cbuf, out);
}